// DGCNN_52347061404124
// MI455X (gfx1250) — compile-verified
//
#include <hip/hip_runtime.h>
#include <hip/hip_bf16.h>

// ---------------------------------------------------------------------------
// DGCNN forward on MI455X (gfx1250, wave32).
// Edge conv collapsed algebraically:  W @ [nbr-ctr; ctr] = Wlo@nbr + (Whi-Wlo)@ctr
//   -> one FP32 WMMA GEMM per layer producing ZZ = X @ [Wlo | Whi-Wlo]^T (N x 2O),
//      then y[n,o] = max_k lrelu((ZZ[idx[n,k],o] + ZZ[n,O+o]) * g*BN + b).
// GEMMs run on V_WMMA_F32_16X16X4_F32 with:
//   * K-pair-major packed weights -> coalesced b64 B-matrix loads
//   * double-buffered 16-K chunks -> next chunk's loads overlap current WMMAs
// ---------------------------------------------------------------------------

typedef __attribute__((ext_vector_type(2))) float v2f;
typedef __attribute__((ext_vector_type(8))) float v8f;

#define NPTS   1024
#define NBATCH 4
#define KSEL   20
#define BN_SCALE 0.99999500003749968f   // 1/sqrt(1+1e-5)
#define NEG_INF  (-__builtin_huge_valf())

// ---------------- pad x (B,N,3) -> xpad (B,N,16), zero-extended ------------
__global__ void pad_kernel(const float* __restrict__ x, float* __restrict__ xp, int total)
{
    int i = blockIdx.x * blockDim.x + threadIdx.x;
    if (i >= total) return;
    float* p = xp + (long)i * 16;
    p[0] = x[i * 3 + 0];
    p[1] = x[i * 3 + 1];
    p[2] = x[i * 3 + 2];
#pragma unroll
    for (int c = 3; c < 16; ++c) p[c] = 0.0f;
}

// ---------------- kNN: one workgroup per (b,n) row -------------------------
// score s_m = 2*x_n.x_m - |x_m|^2 (== d + const per row; same ordering as ref)
__global__ void knn_kernel(const float* __restrict__ X, int ldx, int C,
                           long strideB, int* __restrict__ idxOut, int N)
{
    __shared__ float ctr[256];
    __shared__ float sc[NPTS];
    __shared__ float rv[256];
    __shared__ int   ri[256];

    const int n = blockIdx.x;
    const int b = blockIdx.y;
    const int t = threadIdx.x;
    const float* Xb = X + (long)b * strideB;

    for (int c = t; c < C; c += 256) ctr[c] = Xb[(long)n * ldx + c];
    __syncthreads();

    for (int m = t; m < N; m += 256) {
        const float* xm = Xb + (long)m * ldx;
        float dot = 0.0f, nrm = 0.0f;
        for (int c = 0; c < C; ++c) { float v = xm[c]; dot += ctr[c] * v; nrm += v * v; }
        sc[m] = 2.0f * dot - nrm;
    }
    __syncthreads();

    int* out = idxOut + ((long)b * N + n) * KSEL;
    for (int k = 0; k < KSEL; ++k) {
        float bv = NEG_INF; int bi = N;
        for (int m = t; m < N; m += 256) {
            float v = sc[m];
            if (v > bv || (v == bv && m < bi)) { bv = v; bi = m; }
        }
        rv[t] = bv; ri[t] = bi;
        __syncthreads();
        for (int s = 128; s > 0; s >>= 1) {
            if (t < s) {
                float ov = rv[t + s]; int oi = ri[t + s];
                if (ov > rv[t] || (ov == rv[t] && oi < ri[t])) { rv[t] = ov; ri[t] = oi; }
            }
            __syncthreads();
        }
        if (t == 0) { out[k] = ri[0]; sc[ri[0]] = NEG_INF; }
        __syncthreads();
    }
}

// ---- W (O x 2C) -> packed Weff: logical (col=r in [0,2O), k in [0,Cpad)) ---
// storage: Wp[(k/2)*(2*Nout) + col*2 + (k&1)],  Nout = 2O.
// rows [0,O) = Wlo, rows [O,2O) = Whi - Wlo; zero pad k >= C.
__global__ void weff_kernel(const float* __restrict__ W, float* __restrict__ Wp,
                            int O, int C, int Cpad)
{
    int i = blockIdx.x * blockDim.x + threadIdx.x;
    int total = 2 * O * Cpad;
    if (i >= total) return;
    int c = i % Cpad;              // k index
    int r = i / Cpad;              // output column
    float v = 0.0f;
    if (c < C) {
        if (r < O) v = W[(long)r * 2 * C + c];
        else { int o = r - O; v = W[(long)o * 2 * C + C + c] - W[(long)o * 2 * C + c]; }
    }
    const int Nout = 2 * O;
    Wp[(long)(c >> 1) * (2 * Nout) + r * 2 + (c & 1)] = v;
}

// ---- pack W5 (512 x 960, row-major) into the same K-pair-major layout -----
__global__ void w5pack_kernel(const float* __restrict__ W5, float* __restrict__ Wp,
                              int O /*512*/, int KC /*960*/)
{
    int i = blockIdx.x * blockDim.x + threadIdx.x;
    if (i >= O * KC) return;
    int k = i % KC;
    int o = i / KC;
    Wp[(long)(k >> 1) * (2 * O) + o * 2 + (k & 1)] = W5[i];
}

// ---------------- FP32 WMMA GEMM: Z = X (N x KC) @ Wt^T --------------------
// One wave: 16(M) x 64(N) tile, K chunk = 16 (4 wmma K-steps), double buffered.
// A lane mapping (ISA 7.12.2): lanes 0-15 -> K={0,1}, lanes 16-31 -> K={2,3}.
#define WMMA4(A, Bv, ACC) \
    ACC = __builtin_amdgcn_wmma_f32_16x16x4_f32(false, (A), false, (Bv), (short)0, (ACC), false, false)

#define LOAD_STAGE(AV, BV, K0)                                              \
    {                                                                       \
        _Pragma("unroll")                                                   \
        for (int j = 0; j < 4; ++j) {                                       \
            AV[j] = *(const v2f*)(arow + (K0) + 4 * j);                     \
            const long kb = (long)((K0) + 4 * j + khalf);                   \
            _Pragma("unroll")                                               \
            for (int t = 0; t < 4; ++t)                                     \
                BV[j * 4 + t] = *(const v2f*)(wcol + kb * Nout + t * 32);   \
        }                                                                   \
    }

#define COMPUTE_STAGE(AV, BV)                                               \
    {                                                                       \
        _Pragma("unroll")                                                   \
        for (int j = 0; j < 4; ++j) {                                       \
            WMMA4(AV[j], BV[j * 4 + 0], acc0);                              \
            WMMA4(AV[j], BV[j * 4 + 1], acc1);                              \
            WMMA4(AV[j], BV[j * 4 + 2], acc2);                              \
            WMMA4(AV[j], BV[j * 4 + 3], acc3);                              \
        }                                                                   \
    }

__global__ void __launch_bounds__(32)
gemm16x64_wmma(const float* __restrict__ X, int ldx, long strideXb,
               const float* __restrict__ Wp, int KC, int Nout,
               float* __restrict__ Z, int ldz, long strideZb)
{
    const int lane = threadIdx.x;
    const int m0 = blockIdx.x * 16;
    const int n0 = blockIdx.y * 64;
    const int b  = blockIdx.z;

    const float* Xb = X + (long)b * strideXb;
    float*       Zb = Z + (long)b * strideZb;

    const int l15   = lane & 15;
    const int khalf = (lane >> 4) * 2;               // K offset of this half-wave

    const float* arow = Xb + (long)(m0 + l15) * ldx + khalf;     // A: row-major
    const float* wcol = Wp + (long)(n0 + l15) * 2;               // B: K-pair-major

    v8f acc0 = {}, acc1 = {}, acc2 = {}, acc3 = {};
    v2f aC[4], bC[16], aN[4], bN[16];

    LOAD_STAGE(aC, bC, 0);
    for (int k0 = 0; k0 < KC; k0 += 16) {
        const bool more = (k0 + 16) < KC;
        if (more) { LOAD_STAGE(aN, bN, k0 + 16); }
        COMPUTE_STAGE(aC, bC);
        if (more) {
#pragma unroll
            for (int i = 0; i < 4; ++i)  aC[i] = aN[i];
#pragma unroll
            for (int i = 0; i < 16; ++i) bC[i] = bN[i];
        }
    }

    // D layout: VGPR g, lanes 0-15 -> row m0+g, lanes 16-31 -> row m0+g+8
    const int rbase = m0 + ((lane >> 4) ? 8 : 0);
    const int col   = n0 + l15;
#pragma unroll
    for (int g = 0; g < 8; ++g) {
        float* zr = Zb + (long)(rbase + g) * ldz + col;
        zr[0]  = acc0[g];
        zr[16] = acc1[g];
        zr[32] = acc2[g];
        zr[48] = acc3[g];
    }
}

// ---------------- gather + BN*scale + LeakyReLU + max over k ---------------
__global__ void edgemax_kernel(const float* __restrict__ Z, int ldz, long strideZb,
                               const int* __restrict__ idx,
                               const float* __restrict__ g, const float* __restrict__ bias,
                               float* __restrict__ Y, int ldy, long strideYb,
                               int O, int N)
{
    const int o = blockIdx.x * blockDim.x + threadIdx.x;   // blockDim = 64, O multiple of 64
    const int n = blockIdx.y;
    const int b = blockIdx.z;

    const float* Zb = Z + (long)b * strideZb;
    const int*   id = idx + ((long)b * N + n) * KSEL;

    const float ctr = Zb[(long)n * ldz + O + o];
    const float sc  = g[o] * BN_SCALE;
    const float be  = bias[o];

    float best = NEG_INF;
#pragma unroll 4
    for (int k = 0; k < KSEL; ++k) {
        int m = id[k];
        float v = (Zb[(long)m * ldz + o] + ctr) * sc + be;
        v = (v > 0.0f) ? v : 0.2f * v;
        best = fmaxf(best, v);
    }
    Y[(long)b * strideYb + (long)n * ldy + o] = best;
}

// ---------------- conv5 epilogue: scale + LeakyReLU + max over N -----------
__global__ void pool_kernel(const float* __restrict__ Y, int ldy, long strideYb,
                            const float* __restrict__ g, const float* __restrict__ bias,
                            float* __restrict__ feat, int O, int N)
{
    const int o = blockIdx.x * blockDim.x + threadIdx.x;
    const int b = blockIdx.y;
    const float* Yb = Y + (long)b * strideYb;
    const float sc = g[o] * BN_SCALE;
    const float be = bias[o];
    float best = NEG_INF;
    for (int n = 0; n < N; ++n) {
        float v = Yb[(long)n * ldy + o] * sc + be;
        v = (v > 0.0f) ? v : 0.2f * v;
        best = fmaxf(best, v);
    }
    feat[(long)b * O + o] = best;
}

// ---------------- out = feat (B x 512) @ We^T (256 x 512) ------------------
__global__ void final_kernel(const float* __restrict__ feat, const float* __restrict__ We,
                             float* __restrict__ out, int O, int J)
{
    const int j = blockIdx.x * blockDim.x + threadIdx.x;
    const int b = blockIdx.y;
    if (j >= J) return;
    const float* f = feat + (long)b * O;
    const float* w = We + (long)j * O;
    float acc = 0.0f;
    for (int o = 0; o < O; ++o) acc += f[o] * w[o];
    out[(long)b * J + j] = acc;
}

// ---------------------------------------------------------------------------
extern "C" void kernel_launch(void* const* d_in, const int* in_sizes, int n_in,
                              void* d_out, int out_size, void* d_ws, size_t ws_size,
                              hipStream_t stream)
{
    const float* x  = (const float*)d_in[0];
    const float* W[5]  = { (const float*)d_in[1], (const float*)d_in[4],
                           (const float*)d_in[7], (const float*)d_in[10],
                           (const float*)d_in[13] };
    const float* gv[5] = { (const float*)d_in[2], (const float*)d_in[5],
                           (const float*)d_in[8], (const float*)d_in[11],
                           (const float*)d_in[14] };
    const float* bv[5] = { (const float*)d_in[3], (const float*)d_in[6],
                           (const float*)d_in[9], (const float*)d_in[12],
                           (const float*)d_in[15] };
    const float* We = (const float*)d_in[16];
    float* out = (float*)d_out;

    const int N = NPTS, B = NBATCH;

    // workspace carve-up (bytes)
    char* ws = (char*)d_ws;
    float* xpad = (float*)(ws);                                    // 256 KB : B*N*16
    int*   idx  = (int*)  (ws + 262144);                           // 320 KB : B*N*20
    float* Weff = (float*)(ws + 262144 + 327680);                  //   2 MB : up to 960*512
    float* xc   = (float*)(ws + 262144 + 327680 + 2097152);        //  15 MB : B*N*960
    float* ZZ   = (float*)((char*)xc + (long)B * N * 960 * 4);     //  16 MB : B*N*1024
    float* feat = (float*)((char*)ZZ + (long)B * N * 1024 * 4);    //   8 KB : B*512
    (void)ws_size; (void)n_in; (void)in_sizes; (void)out_size;

    // pad x -> (B,N,16)
    pad_kernel<<<(B * N + 255) / 256, 256, 0, stream>>>(x, xpad, B * N);

    // per-layer config: C_in, padded K, O_out, xc column offset
    const int  Cin[4]   = { 3, 64, 128, 256 };
    const int  Cpad4[4] = { 16, 64, 128, 256 };
    const int  Oout[4]  = { 64, 128, 256, 512 };
    const int  xcoff[4] = { 0, 64, 192, 448 };

    for (int L = 0; L < 4; ++L) {
        const float* src     = (L == 0) ? xpad : (xc + xcoff[L - 1]);
        const int    ldx     = (L == 0) ? 16 : 960;
        const long   strideX = (long)N * ((L == 0) ? 16 : 960);
        const int    O       = Oout[L];
        const int    KC      = Cpad4[L];

        // kNN on input features of this layer
        knn_kernel<<<dim3(N, B), 256, 0, stream>>>(src, ldx, Cin[L], strideX, idx, N);

        // packed Weff = [Wlo ; Whi - Wlo], zero-padded to KC
        {
            int tot = 2 * O * KC;
            weff_kernel<<<(tot + 255) / 256, 256, 0, stream>>>(W[L], Weff, O, Cin[L], KC);
        }

        // ZZ (N x 2O) = X @ Weff^T   -- FP32 WMMA
        gemm16x64_wmma<<<dim3(N / 16, (2 * O) / 64, B), 32, 0, stream>>>(
            src, ldx, strideX, Weff, KC, 2 * O, ZZ, 2 * O, (long)N * 2 * O);

        // gather neighbors, LeakyReLU, max over k -> write into concat buffer
        edgemax_kernel<<<dim3(O / 64, N, B), 64, 0, stream>>>(
            ZZ, 2 * O, (long)N * 2 * O, idx, gv[L], bv[L],
            xc + xcoff[L], 960, (long)N * 960, O, N);
    }

    // conv5: pack W5 into K-pair-major, then Y5 (N x 512) = xc @ W5^T
    {
        int tot = 512 * 960;
        w5pack_kernel<<<(tot + 255) / 256, 256, 0, stream>>>(W[4], Weff, 512, 960);
    }
    gemm16x64_wmma<<<dim3(N / 16, 512 / 64, B), 32, 0, stream>>>(
        xc, 960, (long)N * 960, Weff, 960, 512, ZZ, 512, (long)N * 512);

    // scale + LeakyReLU + global max over N -> feat (B x 512)
    pool_kernel<<<dim3(512 / 256, B), 256, 0, stream>>>(
        ZZ, 512, (long)N * 512, gv[4], bv[4], feat, 512, N);

    // out (B x 256) = feat @ We^T
    final_kernel<<<dim3(1, B), 256, 0, stream>>>(feat, We, out, 512, 256);
}